// NaiveVisCache_28192165331586
// MI455X (gfx1250) — compile-verified
//
#include <hip/hip_runtime.h>
#include <stdint.h>

// ---------------------------------------------------------------------------
// NaiveVisCache gather kernel for MI455X (gfx1250, wave32).
//
// Memory strategy (MI455X: 23.3 TB/s HBM, 192 MB L2):
//  * numer+denom tables = 100.6 MB  -> resident in L2; gathers use RT policy.
//  * ray streams (112 MB) use NT loads/stores so they never evict the tables.
//  * ray input is staged through LDS by the Tensor Data Mover (async, double
//    buffered, synchronized with s_wait_tensorcnt) so the vector memory pipe
//    is dedicated to the random 32-bit gathers.
// ---------------------------------------------------------------------------

typedef float    f32x4 __attribute__((ext_vector_type(4)));
typedef unsigned u32x4 __attribute__((ext_vector_type(4)));
typedef int      i32x4 __attribute__((ext_vector_type(4)));
typedef int      i32x8 __attribute__((ext_vector_type(8)));

#define GRID_N          128
#define BLOCK_THREADS   256
#define WAVES_PER_BLOCK 8
#define RAYS_PER_WAVE   128                  // 32 lanes * 4 rays
#define TILE_BYTES      (RAYS_PER_WAVE * 12) // 1536 B per stream (xyz f32)
#define TILE_QW         (TILE_BYTES / 8)     // 192 8-byte elements (TDM units)
#define TILE_V4         (TILE_BYTES / 16)    // 96 float4

#if defined(__AMDGCN__) && __has_builtin(__builtin_amdgcn_tensor_load_to_lds) && \
    __has_builtin(__builtin_amdgcn_s_wait_tensorcnt)
#define USE_TDM 1
#else
#define USE_TDM 0
#endif

#if USE_TDM
// 1-D TDM copy of TILE_BYTES contiguous bytes (as 192 x 8B) global -> LDS.
// D# layout per CDNA5 ISA ch.8 (CDNA5 ISA):
//  group0: [1:0]=count=1, [63:32]=lds_addr, [120:64]=global_addr, [127:126]=2
//  group1: [17:16]=data_size(3=8B), tensor_dim0, tensor_dim1=1,
//          tile_dim0=192, tile_dim1=1, tensor_dim0_stride=192
// This toolchain's builtin takes 6 operands (g0 x4, g1 x8, g2 x4, g3 x4,
// extra x8, cpol); trailing groups are zero (unused: 1-D tile, tile_dim2=0).
__device__ __forceinline__ void tdm_load_tile(uint32_t lds_byte_off,
                                              uint64_t gaddr) {
  u32x4 g0;
  g0[0] = 1u;                                                     // count=1
  g0[1] = lds_byte_off;                                           // lds_addr
  g0[2] = (uint32_t)gaddr;                                        // addr[31:0]
  g0[3] = (uint32_t)((gaddr >> 32) & 0x01FFFFFFu) | 0x80000000u;  // addr[56:32]|type=2
  i32x8 g1;
  g1[0] = (int)(3u << 16);                        // data_size = 8 bytes
  g1[1] = (int)((TILE_QW & 0xFFFF) << 16);        // tensor_dim0[15:0]
  g1[2] = (int)((TILE_QW >> 16) | (1 << 16));     // tensor_dim0[31:16] | tensor_dim1=1
  g1[3] = (int)(TILE_QW << 16);                   // tile_dim0 = 192
  g1[4] = 1;                                      // tile_dim1 = 1
  g1[5] = TILE_QW;                                // tensor_dim0_stride
  g1[6] = 0;
  g1[7] = 0;
  i32x4 z4 = {0, 0, 0, 0};
  i32x8 z8 = {0, 0, 0, 0, 0, 0, 0, 0};
  __builtin_amdgcn_tensor_load_to_lds(g0, g1, z4, z4, z8, 0);
}
#endif

__global__ __launch_bounds__(BLOCK_THREADS) void viscache_kernel(
    const float* __restrict__ org, const float* __restrict__ dir,
    const int* __restrict__ numer, const int* __restrict__ denom,
    float* __restrict__ out, int nrays, int nTiles, int waveStride) {
#if USE_TDM
  // 8 waves * 2 buffers * (org tile + dir tile) = 48 KB (of 320 KB/WGP)
  __shared__ f32x4 smem[WAVES_PER_BLOCK * 2 * 2 * TILE_V4];
#endif
  const int wave = threadIdx.x >> 5;
  const int lane = threadIdx.x & 31;
  int tile = blockIdx.x * WAVES_PER_BLOCK + wave;
  if (tile >= nTiles) return;

#if USE_TDM
  f32x4* wbase = smem + wave * (4 * TILE_V4);
  const uint32_t ldsBase = (uint32_t)(uintptr_t)wbase;  // flat addr low 32 = LDS offset
  const uint64_t orgA = (uint64_t)(uintptr_t)org;
  const uint64_t dirA = (uint64_t)(uintptr_t)dir;

  bool curIssued = false;
  {
    long long rb = (long long)tile * RAYS_PER_WAVE;
    if (rb + RAYS_PER_WAVE <= (long long)nrays) {
      tdm_load_tile(ldsBase, orgA + (uint64_t)rb * 12u);
      tdm_load_tile(ldsBase + TILE_BYTES, dirA + (uint64_t)rb * 12u);
      curIssued = true;
    }
  }
  int cur = 0;
#endif

  for (; tile < nTiles; tile += waveStride) {
    const long long rayBase = (long long)tile * RAYS_PER_WAVE;
    const bool full = (rayBase + RAYS_PER_WAVE) <= (long long)nrays;

    float o[12] __attribute__((aligned(16)));
    float v[12] __attribute__((aligned(16)));
    bool haveVec = false;

#if USE_TDM
    // Issue the DMA for the next tile into the other buffer before waiting.
    bool nextIssued = false;
    {
      const int nt = tile + waveStride;
      if (nt < nTiles) {
        const long long nrb = (long long)nt * RAYS_PER_WAVE;
        if (nrb + RAYS_PER_WAVE <= (long long)nrays) {
          const uint32_t lo = ldsBase + (uint32_t)(cur ^ 1) * (2 * TILE_BYTES);
          tdm_load_tile(lo, orgA + (uint64_t)nrb * 12u);
          tdm_load_tile(lo + TILE_BYTES, dirA + (uint64_t)nrb * 12u);
          nextIssued = true;
        }
      }
    }
    if (curIssued) {
      if (nextIssued) __builtin_amdgcn_s_wait_tensorcnt(2);  // allow next tile in flight
      else            __builtin_amdgcn_s_wait_tensorcnt(0);
    }
    if (full && curIssued) {
      const f32x4* po = wbase + cur * (2 * TILE_V4) + lane * 3;  // 48 B / lane
      const f32x4* pv = po + TILE_V4;
      *(f32x4*)&o[0] = po[0]; *(f32x4*)&o[4] = po[1]; *(f32x4*)&o[8] = po[2];
      *(f32x4*)&v[0] = pv[0]; *(f32x4*)&v[4] = pv[1]; *(f32x4*)&v[8] = pv[2];
      haveVec = true;
    }
#else
    if (full) {
      const f32x4* po = (const f32x4*)(org + rayBase * 3) + lane * 3;
      const f32x4* pv = (const f32x4*)(dir + rayBase * 3) + lane * 3;
      // NT: streaming data must not evict the L2-resident vis tables.
      *(f32x4*)&o[0] = __builtin_nontemporal_load(po + 0);
      *(f32x4*)&o[4] = __builtin_nontemporal_load(po + 1);
      *(f32x4*)&o[8] = __builtin_nontemporal_load(po + 2);
      *(f32x4*)&v[0] = __builtin_nontemporal_load(pv + 0);
      *(f32x4*)&v[4] = __builtin_nontemporal_load(pv + 1);
      *(f32x4*)&v[8] = __builtin_nontemporal_load(pv + 2);
      haveVec = true;
      const long long nrb = (long long)(tile + waveStride) * RAYS_PER_WAVE;
      if (nrb < (long long)nrays) {          // global_prefetch_b8 next tile
        __builtin_prefetch(org + nrb * 3 + lane * 12, 0, 0);
        __builtin_prefetch(dir + nrb * 3 + lane * 12, 0, 0);
      }
    }
#endif

    if (!haveVec) {  // ragged tail: per-ray direct loads
#pragma unroll
      for (int p = 0; p < 4; ++p) {
        const long long r = rayBase + lane * 4 + p;
        if (r < (long long)nrays) {
          o[3*p+0] = org[r*3+0]; o[3*p+1] = org[r*3+1]; o[3*p+2] = org[r*3+2];
          v[3*p+0] = dir[r*3+0]; v[3*p+1] = dir[r*3+1]; v[3*p+2] = dir[r*3+2];
        } else {
          o[3*p+0] = o[3*p+1] = o[3*p+2] = 0.0f;
          v[3*p+0] = 1.0f; v[3*p+1] = 0.0f; v[3*p+2] = 0.0f;
        }
      }
    }

    f32x4 res;
#pragma unroll
    for (int p = 0; p < 4; ++p) {
      const float dx = v[3*p+0], dy = v[3*p+1], dz = v[3*p+2];
      const float m = fmaxf(fabsf(dx), fmaxf(fabsf(dy), fabsf(dz)));
      // d/m >= 1  <=>  d >= m (m>0, RTE): same override order as reference.
      int face = 0;
      face = (dx <= -m) ? 1 : face;
      face = (dy >=  m) ? 2 : face;
      face = (dy <= -m) ? 3 : face;
      face = (dz >=  m) ? 4 : face;
      face = (dz <= -m) ? 5 : face;
      const int ci = (int)fminf(fmaxf((o[3*p+0]*0.5f + 0.5f)*127.0f, 0.0f), 127.0f);
      const int cj = (int)fminf(fmaxf((o[3*p+1]*0.5f + 0.5f)*127.0f, 0.0f), 127.0f);
      const int ck = (int)fminf(fmaxf((o[3*p+2]*0.5f + 0.5f)*127.0f, 0.0f), 127.0f);
      const int idx = ci * (GRID_N * GRID_N * 6) + cj * (GRID_N * 6) + ck * 6 + face;
      const int n = numer[idx];   // RT policy: stays L2-resident (100 MB < 192 MB)
      const int d = denom[idx];
      res[p] = (float)n / (float)d;
    }

    const long long outBase = rayBase + lane * 4;
    if (outBase + 4 <= (long long)nrays) {
      __builtin_nontemporal_store(res, (f32x4*)(out + outBase));
    } else {
#pragma unroll
      for (int p = 0; p < 4; ++p)
        if (outBase + p < (long long)nrays) out[outBase + p] = res[p];
    }

#if USE_TDM
    cur ^= 1;
    curIssued = nextIssued;
#endif
  }
}

extern "C" void kernel_launch(void* const* d_in, const int* in_sizes, int n_in,
                              void* d_out, int out_size, void* d_ws, size_t ws_size,
                              hipStream_t stream) {
  const float* org   = (const float*)d_in[0];
  const float* dir   = (const float*)d_in[1];
  const int*   numer = (const int*)d_in[2];
  const int*   denom = (const int*)d_in[3];
  float*       out   = (float*)d_out;

  const int nrays  = in_sizes[0] / 3;                       // 4,194,304
  const int nTiles = (nrays + RAYS_PER_WAVE - 1) / RAYS_PER_WAVE;  // 32768

  int blocks = (nTiles + WAVES_PER_BLOCK - 1) / WAVES_PER_BLOCK;
  if (blocks > 1024) blocks = 1024;   // 8192 waves, ~4 tiles each -> pipelined
  const int waveStride = blocks * WAVES_PER_BLOCK;

  viscache_kernel<<<blocks, BLOCK_THREADS, 0, stream>>>(
      org, dir, numer, denom, out, nrays, nTiles, waveStride);
}